// GraphNeuralNetwork_11742440587994
// MI455X (gfx1250) — compile-verified
//
#include <hip/hip_runtime.h>
#include <hip/hip_bf16.h>

typedef float v2f __attribute__((ext_vector_type(2)));
typedef float v8f __attribute__((ext_vector_type(8)));

#define BGRAPHS 64
#define EDGES   1048576
#define EPSV    1e-5f

// ---------------------------------------------------------------------------
// GEMM: C[n,N] = A[n,K] * B[K,N], fp32 WMMA 16x16x4, one 16x16 tile per wave.
// A layout (16x4 f32): lanes 0-15 hold row M=lane, {K0,K1}; lanes 16-31 {K2,K3}
// B layout (4x16 f32): lanes 0-15 hold col N=lane, {K0,K1}; lanes 16-31 {K2,K3}
// C/D layout: lanes 0-15 col=lane rows 0-7 (vgpr r -> row r); lanes 16-31 rows 8-15
// ---------------------------------------------------------------------------
__global__ void __launch_bounds__(128)
gemm_f32_wmma(const float* __restrict__ A, const float* __restrict__ Bm,
              float* __restrict__ C, int n, int K, int N) {
    const int lane = threadIdx.x;                                   // 0..31
    const int wave = blockIdx.x * blockDim.y + threadIdx.y;
    const int tilesN = N >> 4;
    const int tm = (wave / tilesN) << 4;
    const int tn = (wave % tilesN) << 4;
    const int row = lane & 15;
    const int kb  = (lane >> 4) << 1;                               // 0 or 2

    v8f c = {};
    const float* Arow = A + (size_t)(tm + row) * K;
    const float* Bcol = Bm + (tn + row);
    for (int k0 = 0; k0 < K; k0 += 4) {
        v2f a, b;
        a.x = Arow[k0 + kb];
        a.y = Arow[k0 + kb + 1];
        b.x = Bcol[(size_t)(k0 + kb) * N];
        b.y = Bcol[(size_t)(k0 + kb + 1) * N];
        c = __builtin_amdgcn_wmma_f32_16x16x4_f32(false, a, false, b,
                                                  (short)0, c, false, false);
    }
    const int ccol  = lane & 15;
    const int rbase = (lane >> 4) << 3;
    float* Cout = C + (size_t)(tm + rbase) * N + tn + ccol;
#pragma unroll
    for (int r = 0; r < 8; ++r) Cout[(size_t)r * N] = c[r];
}

// ---------------------------------------------------------------------------
__global__ void init_edges(const int* __restrict__ ei, int* srcW, int* dstW,
                           int* maskW, int E_) {
    int e = blockIdx.x * blockDim.x + threadIdx.x;
    if (e < E_) { srcW[e] = ei[e]; dstW[e] = ei[E_ + e]; maskW[e] = 1; }
}

__global__ void fill_f32(float* p, float v, int n) {
    int i = blockIdx.x * blockDim.x + threadIdx.x;
    if (i < n) p[i] = v;
}
__global__ void fill_i32(int* p, int v, int n) {
    int i = blockIdx.x * blockDim.x + threadIdx.x;
    if (i < n) p[i] = v;
}

__global__ void deg_accum(const int* __restrict__ dstW, const int* __restrict__ maskW,
                          float* deg, int E_) {
    int e = blockIdx.x * blockDim.x + threadIdx.x;
    if (e < E_ && maskW[e]) atomicAdd(deg + dstW[e], 1.0f);
}

__global__ void compute_dinv(const float* __restrict__ deg, float* dinv, int n) {
    int v = blockIdx.x * blockDim.x + threadIdx.x;
    if (v < n) dinv[v] = rsqrtf(deg[v]);
}

// acc = h/deg + bias  (self-loop term + bias, also initializes accumulator)
__global__ void self_bias(const float* __restrict__ h, const float* __restrict__ deg,
                          const float* __restrict__ b, float* acc, int n, int dout) {
    int idx = blockIdx.x * blockDim.x + threadIdx.x;
    int total = n * dout;
    if (idx < total) {
        int v = idx / dout, f = idx - v * dout;
        acc[idx] = h[idx] / deg[v] + b[f];
    }
}

// one block per edge, blockDim = dout; acc[dst] += h[src] * dinv[src]*dinv[dst]
__global__ void edge_scatter(const int* __restrict__ srcW, const int* __restrict__ dstW,
                             const int* __restrict__ maskW, const float* __restrict__ dinv,
                             const float* __restrict__ h, float* acc, int dout) {
    int e = blockIdx.x;
    if (!maskW[e]) return;
    int s = srcW[e], d = dstW[e];
    float w = dinv[s] * dinv[d];
    int f = threadIdx.x;
    atomicAdd(&acc[(size_t)d * dout + f], h[(size_t)s * dout + f] * w);
}

// GraphNorm + ReLU, in place. One block per graph, one thread per feature.
__global__ void graphnorm_relu(float* x, const float* __restrict__ w,
                               const float* __restrict__ bias, const float* __restrict__ ms,
                               int nper, int dout) {
    int g = blockIdx.x, f = threadIdx.x;
    const size_t base = (size_t)g * nper * dout + f;
    float sum = 0.f;
    for (int j = 0; j < nper; ++j) sum += x[base + (size_t)j * dout];
    float mean = sum / (float)nper;
    float msf = ms[f];
    float sum2 = 0.f;
    for (int j = 0; j < nper; ++j) {
        float s = x[base + (size_t)j * dout] - msf * mean;
        sum2 += s * s;
    }
    float inv = rsqrtf(sum2 / (float)nper + EPSV);
    float wf = w[f], bf = bias[f];
    for (int j = 0; j < nper; ++j) {
        float s = x[base + (size_t)j * dout] - msf * mean;
        float y = wf * s * inv + bf;
        x[base + (size_t)j * dout] = y > 0.f ? y : 0.f;
    }
}

__global__ void pnorm_inv(const float* __restrict__ p, float* out, int dout) {
    __shared__ float red[256];
    int t = threadIdx.x;
    float v = (t < dout) ? p[t] * p[t] : 0.f;
    red[t] = v;
    __syncthreads();
    for (int s = 128; s > 0; s >>= 1) {
        if (t < s) red[t] += red[t + s];
        __syncthreads();
    }
    if (t == 0) *out = rsqrtf(red[0]);
}

// one wave per node: score = tanh((x . p) / ||p||)
__global__ void node_score(const float* __restrict__ x, const float* __restrict__ p,
                           const float* __restrict__ invn, float* score, int n, int dout) {
    int wid = (blockIdx.x * blockDim.x + threadIdx.x) >> 5;
    int lane = threadIdx.x & 31;
    if (wid >= n) return;
    const float* xr = x + (size_t)wid * dout;
    float dot = 0.f;
    for (int f = lane; f < dout; f += 32) dot += xr[f] * p[f];
    for (int off = 16; off > 0; off >>= 1) dot += __shfl_down(dot, off, 32);
    if (lane == 0) score[wid] = tanhf(dot * (*invn));
}

// one block per graph, blockDim = nper (<=1024). Bitonic sort (score desc, idx asc),
// then emit the top kkeep: their tanh values, source node ids, and new-id mapping.
__global__ void __launch_bounds__(1024)
topk_select(const float* __restrict__ score, float* selval, int* selidx,
            int* mapping, int nper, int kkeep) {
    __shared__ float sk[1024];
    __shared__ int   si[1024];
    int g = blockIdx.x, t = threadIdx.x;
    sk[t] = score[g * nper + t];
    si[t] = t;
    __syncthreads();
    for (int ksz = 2; ksz <= nper; ksz <<= 1) {
        for (int j = ksz >> 1; j > 0; j >>= 1) {
            int ixj = t ^ j;
            if (ixj > t) {
                float s1 = sk[t], s2 = sk[ixj];
                int i1 = si[t], i2 = si[ixj];
                bool beforeT = (s1 > s2) || (s1 == s2 && i1 < i2);
                bool dirUp = ((t & ksz) == 0);
                if (dirUp ? !beforeT : beforeT) {
                    sk[t] = s2; sk[ixj] = s1;
                    si[t] = i2; si[ixj] = i1;
                }
            }
            __syncthreads();
        }
    }
    if (t < kkeep) {
        int old = si[t];
        selval[g * kkeep + t]        = sk[t];
        selidx[g * kkeep + t]        = g * nper + old;
        mapping[g * nper + old]      = g * kkeep + t;
    }
}

__global__ void gather_scale(const float* __restrict__ xsrc, const float* __restrict__ selval,
                             const int* __restrict__ selidx, float* xdst, int nnew, int dout) {
    int idx = blockIdx.x * blockDim.x + threadIdx.x;
    int total = nnew * dout;
    if (idx >= total) return;
    int u = idx / dout, f = idx - u * dout;
    xdst[idx] = xsrc[(size_t)selidx[u] * dout + f] * selval[u];
}

__global__ void edge_remap(int* srcW, int* dstW, int* maskW,
                           const int* __restrict__ mapping, int E_) {
    int e = blockIdx.x * blockDim.x + threadIdx.x;
    if (e >= E_) return;
    if (maskW[e]) {
        int ns = mapping[srcW[e]], nd = mapping[dstW[e]];
        bool ok = (ns >= 0) && (nd >= 0);
        srcW[e]  = ok ? ns : 0;
        dstW[e]  = ok ? nd : 0;
        maskW[e] = ok ? 1 : 0;
    }
}

// one block per graph, blockDim = dout(=256): concat(mean, max) over nodes
__global__ void global_pool(const float* __restrict__ x, float* out, int nper, int dout) {
    int g = blockIdx.x, f = threadIdx.x;
    const float* xb = x + (size_t)g * nper * dout + f;
    float sum = 0.f, mx = -__builtin_inff();
    for (int j = 0; j < nper; ++j) {
        float v = xb[(size_t)j * dout];
        sum += v;
        mx = fmaxf(mx, v);
    }
    out[(size_t)g * 2 * dout + f]        = sum / (float)nper;
    out[(size_t)g * 2 * dout + dout + f] = mx;
}

// ---------------------------------------------------------------------------
extern "C" void kernel_launch(void* const* d_in, const int* in_sizes, int n_in,
                              void* d_out, int out_size, void* d_ws, size_t ws_size,
                              hipStream_t stream) {
    (void)in_sizes; (void)n_in; (void)out_size; (void)ws_size;

    const float* x0 = (const float*)d_in[0];
    const int*   ei = (const int*)d_in[1];
    const float* W[3]   = {(const float*)d_in[3],  (const float*)d_in[9],  (const float*)d_in[15]};
    const float* bb[3]  = {(const float*)d_in[4],  (const float*)d_in[10], (const float*)d_in[16]};
    const float* gw[3]  = {(const float*)d_in[5],  (const float*)d_in[11], (const float*)d_in[17]};
    const float* gbi[3] = {(const float*)d_in[6],  (const float*)d_in[12], (const float*)d_in[18]};
    const float* gms[3] = {(const float*)d_in[7],  (const float*)d_in[13], (const float*)d_in[19]};
    const float* pp[3]  = {(const float*)d_in[8],  (const float*)d_in[14], (const float*)d_in[20]};

    // Workspace layout (256B aligned slabs)
    char* wsb = (char*)d_ws;
    size_t off = 0;
    auto take = [&](size_t bytes) {
        char* p = wsb + off;
        off += (bytes + 255) & ~(size_t)255;
        return p;
    };
    float* bufH   = (float*)take((size_t)8 * 1024 * 1024 * 4);   // GEMM out (max 8M f32)
    float* bufAcc = (float*)take((size_t)8 * 1024 * 1024 * 4);   // GCN accum / normed x
    float* bufX   = (float*)take((size_t)4 * 1024 * 1024 * 4);   // pooled node features
    float* deg    = (float*)take(65536 * 4);
    float* dinv   = (float*)take(65536 * 4);
    float* score  = (float*)take(65536 * 4);
    int*   mapping= (int*)  take(65536 * 4);
    int*   srcW   = (int*)  take((size_t)EDGES * 4);
    int*   dstW   = (int*)  take((size_t)EDGES * 4);
    int*   maskW  = (int*)  take((size_t)EDGES * 4);
    float* selval = (float*)take(32768 * 4);
    int*   selidx = (int*)  take(32768 * 4);
    float* pinv   = (float*)take(256);

    const int dins[3]  = {100, 128, 128};
    const int douts[3] = {128, 128, 256};

    init_edges<<<EDGES / 256, 256, 0, stream>>>(ei, srcW, dstW, maskW, EDGES);

    int nper = 1024;
    const float* xin = x0;
    for (int i = 0; i < 3; ++i) {
        const int n = BGRAPHS * nper;
        const int dk = dins[i], dout = douts[i];

        // h = x @ W  (fp32 WMMA)
        const int tiles = (n / 16) * (dout / 16);
        gemm_f32_wmma<<<dim3(tiles / 4), dim3(32, 4), 0, stream>>>(xin, W[i], bufH, n, dk, dout);

        // degree (with +1 self loop) and D^-1/2
        fill_f32<<<(n + 255) / 256, 256, 0, stream>>>(deg, 1.0f, n);
        deg_accum<<<(EDGES + 255) / 256, 256, 0, stream>>>(dstW, maskW, deg, EDGES);
        compute_dinv<<<(n + 255) / 256, 256, 0, stream>>>(deg, dinv, n);

        // acc = h/deg + b, then scatter normalized neighbor messages
        const int total = n * dout;
        self_bias<<<(total + 255) / 256, 256, 0, stream>>>(bufH, deg, bb[i], bufAcc, n, dout);
        edge_scatter<<<dim3(EDGES), dim3(dout), 0, stream>>>(srcW, dstW, maskW, dinv,
                                                             bufH, bufAcc, dout);

        // GraphNorm + ReLU (in place on bufAcc)
        graphnorm_relu<<<dim3(BGRAPHS), dim3(dout), 0, stream>>>(bufAcc, gw[i], gbi[i],
                                                                 gms[i], nper, dout);

        // TopK pooling
        pnorm_inv<<<1, 256, 0, stream>>>(pp[i], pinv, dout);
        node_score<<<(n * 32) / 256, 256, 0, stream>>>(bufAcc, pp[i], pinv, score, n, dout);
        fill_i32<<<(n + 255) / 256, 256, 0, stream>>>(mapping, -1, n);
        const int kkeep = nper / 2;
        topk_select<<<dim3(BGRAPHS), dim3(nper), 0, stream>>>(score, selval, selidx,
                                                              mapping, nper, kkeep);
        const int nnew = BGRAPHS * kkeep;
        gather_scale<<<(nnew * dout + 255) / 256, 256, 0, stream>>>(bufAcc, selval, selidx,
                                                                    bufX, nnew, dout);
        edge_remap<<<(EDGES + 255) / 256, 256, 0, stream>>>(srcW, dstW, maskW, mapping, EDGES);

        nper = kkeep;
        xin = bufX;
    }

    // final n_per = 128, dout = 256 -> [64, 512]
    global_pool<<<dim3(BGRAPHS), dim3(256), 0, stream>>>(bufX, (float*)d_out, nper, 256);
}